// ADV_75574244541027
// MI455X (gfx1250) — compile-verified
//
#include <hip/hip_runtime.h>
#include <math.h>

typedef float v2f __attribute__((ext_vector_type(2)));
typedef float v8f __attribute__((ext_vector_type(8)));

#define BIAS  1e-4f
#define TAU   0.2f
#define TEMP  0.2f
#define BDIM  2048
#define DDIM  64
#define ROWS  8192            // K*B = 4*2048
#define NST   256             // 8192/32 supertiles per dim

// workspace layout (floats)
#define WS_ACC 0              // [0]=sum_add [1]=sum_drop [2]=cnt0 [3]=cnt1
#define WS_ROW 8              // 8192 row sums of exp(sim)
#define WS_COL (WS_ROW + ROWS)
#define WS_POS (WS_COL + ROWS) // 8192 diag entries
#define WS_A   (WS_POS + ROWS) // 8192*64 normalized z1 (1/TEMP folded in)
#define WS_B   (WS_A + ROWS*DDIM)

// ---------------- zero accumulators ----------------
__global__ void ADV_k_zero(float* ws, int n) {
    for (int i = blockIdx.x * blockDim.x + threadIdx.x; i < n;
         i += gridDim.x * blockDim.x)
        ws[i] = 0.0f;
}

// ---------------- reg_aug streaming reduction ----------------
__device__ __forceinline__ void ADV_proc(float xv, float pv, float uv,
                                         float& sA, float& sD,
                                         float& c0, float& c1) {
    float delta = 1.0f - BIAS - (1.0f - 2.0f * BIAS) * uv;
    float logit = __logf(delta) - log1pf(-delta) + pv;
    float g = 1.0f / (1.0f + __expf(-logit * (1.0f / TAU)));
    float t = fabsf(1.0f - g);           // == |aug| (x==0) and |1-aug| (x==1)
    float m1 = (xv == 1.0f) ? 1.0f : 0.0f;
    sD += t * m1;
    sA += t * (1.0f - m1);
    c1 += m1;
    c0 += 1.0f - m1;
}

__global__ void __launch_bounds__(256)
ADV_k_reg(const float* __restrict__ x, const float* __restrict__ p,
          const float* __restrict__ u, float* ws, int n) {
    float sA = 0.f, sD = 0.f, c0 = 0.f, c1 = 0.f;
    int tid = blockIdx.x * blockDim.x + threadIdx.x;
    int stride = gridDim.x * blockDim.x;
    int n4 = n >> 2;
    const float4* x4 = (const float4*)x;
    const float4* p4 = (const float4*)p;
    const float4* u4 = (const float4*)u;
    for (int i = tid; i < n4; i += stride) {
        float4 xv = x4[i], pv = p4[i], uv = u4[i];
        ADV_proc(xv.x, pv.x, uv.x, sA, sD, c0, c1);
        ADV_proc(xv.y, pv.y, uv.y, sA, sD, c0, c1);
        ADV_proc(xv.z, pv.z, uv.z, sA, sD, c0, c1);
        ADV_proc(xv.w, pv.w, uv.w, sA, sD, c0, c1);
    }
    for (int i = (n4 << 2) + tid; i < n; i += stride)
        ADV_proc(x[i], p[i], u[i], sA, sD, c0, c1);

    // wave32 reduction, lane 0 of each wave does the atomics
    for (int off = 16; off; off >>= 1) {
        sA += __shfl_down(sA, off);
        sD += __shfl_down(sD, off);
        c0 += __shfl_down(c0, off);
        c1 += __shfl_down(c1, off);
    }
    if ((threadIdx.x & 31) == 0) {
        atomicAdd(&ws[WS_ACC + 0], sA);
        atomicAdd(&ws[WS_ACC + 1], sD);
        atomicAdd(&ws[WS_ACC + 2], c0);
        atomicAdd(&ws[WS_ACC + 3], c1);
    }
}

// ---------------- normalize rows + diagonal ----------------
__global__ void ADV_k_norm(const float* __restrict__ z1,
                           const float* __restrict__ z2, float* ws) {
    int row = blockIdx.x * blockDim.x + threadIdx.x;
    if (row >= ROWS) return;
    const float* a = z1 + row * DDIM;
    const float* b = z2 + row * DDIM;
    float n1 = 0.f, n2 = 0.f, dp = 0.f;
#pragma unroll
    for (int d = 0; d < DDIM; ++d) {
        float av = a[d], bv = b[d];
        n1 += av * av;
        n2 += bv * bv;
        dp += av * bv;
    }
    float i1 = rsqrtf(n1), i2 = rsqrtf(n2);
    float sA = i1 * (1.0f / TEMP);       // fold 1/TEMP into A side
    float* Ah = ws + WS_A + row * DDIM;
    float* Bh = ws + WS_B + row * DDIM;
#pragma unroll
    for (int d = 0; d < DDIM; ++d) {
        Ah[d] = a[d] * sA;
        Bh[d] = b[d] * i2;
    }
    ws[WS_POS + row] = __expf(dp * i1 * i2 * (1.0f / TEMP));
}

// ---------------- gram + exp + row/col sums (WMMA f32 16x16x4) ----------------
__global__ void __launch_bounds__(256)
ADV_k_gram(float* ws) {
    const float* __restrict__ A  = ws + WS_A;
    const float* __restrict__ Bm = ws + WS_B;
    float* rowS = ws + WS_ROW;
    float* colS = ws + WS_COL;

    int wave = threadIdx.x >> 5;
    int lane = threadIdx.x & 31;
    int st = blockIdx.x * 8 + wave;               // supertile id
    int rb = (st / NST) * 32;                     // row base
    int cb = (st % NST) * 32;                     // col base
    int l15 = lane & 15;
    int half = lane >> 4;
    int kofs = half * 2;                          // lanes 16-31 hold K+2,K+3

    const float* a0p = A  + (rb + l15)      * DDIM + kofs;
    const float* a1p = A  + (rb + 16 + l15) * DDIM + kofs;
    const float* b0p = Bm + (cb + l15)      * DDIM + kofs;
    const float* b1p = Bm + (cb + 16 + l15) * DDIM + kofs;

    v8f c00 = {}, c01 = {}, c10 = {}, c11 = {};
#pragma unroll
    for (int kk = 0; kk < DDIM; kk += 4) {
        v2f a0 = {a0p[kk], a0p[kk + 1]};
        v2f a1 = {a1p[kk], a1p[kk + 1]};
        v2f b0 = {b0p[kk], b0p[kk + 1]};
        v2f b1 = {b1p[kk], b1p[kk + 1]};
        c00 = __builtin_amdgcn_wmma_f32_16x16x4_f32(false, a0, false, b0,
                                                    (short)0, c00, false, false);
        c01 = __builtin_amdgcn_wmma_f32_16x16x4_f32(false, a0, false, b1,
                                                    (short)0, c01, false, false);
        c10 = __builtin_amdgcn_wmma_f32_16x16x4_f32(false, a1, false, b0,
                                                    (short)0, c10, false, false);
        c11 = __builtin_amdgcn_wmma_f32_16x16x4_f32(false, a1, false, b1,
                                                    (short)0, c11, false, false);
    }

    // exp + reductions. C layout: VGPR r, lanes 0-15 -> (row r, col l15),
    // lanes 16-31 -> (row r+8, col l15).
    float e00[8], e01[8], e10[8], e11[8];
    float cp0 = 0.f, cp1 = 0.f;
#pragma unroll
    for (int r = 0; r < 8; ++r) {
        e00[r] = __expf(c00[r]);
        e01[r] = __expf(c01[r]);
        e10[r] = __expf(c10[r]);
        e11[r] = __expf(c11[r]);
        cp0 += e00[r] + e10[r];   // cols cb..cb+15
        cp1 += e01[r] + e11[r];   // cols cb+16..cb+31
    }
    atomicAdd(&colS[cb + l15], cp0);
    atomicAdd(&colS[cb + 16 + l15], cp1);

#pragma unroll
    for (int r = 0; r < 8; ++r) {
        float v0 = e00[r] + e01[r];   // rows rb + (half? r+8 : r)
        float v1 = e10[r] + e11[r];   // rows rb+16 + (half? r+8 : r)
#pragma unroll
        for (int m = 1; m < 16; m <<= 1) {
            v0 += __shfl_xor(v0, m);
            v1 += __shfl_xor(v1, m);
        }
        if (l15 == 0) {
            atomicAdd(&rowS[rb + (half << 3) + r], v0);
            atomicAdd(&rowS[rb + 16 + (half << 3) + r], v1);
        }
    }
}

// ---------------- final combine ----------------
__global__ void ADV_k_final(const float* __restrict__ ws, float* out) {
    __shared__ float s0[256], s1[256];
    float l0 = 0.f, l1 = 0.f;
    for (int i = threadIdx.x; i < ROWS; i += 256) {
        float d = ws[WS_POS + i];
        l0 += __logf(d / (ws[WS_COL + i] - d));
        l1 += __logf(d / (ws[WS_ROW + i] - d));
    }
    s0[threadIdx.x] = l0;
    s1[threadIdx.x] = l1;
    __syncthreads();
    for (int off = 128; off; off >>= 1) {
        if (threadIdx.x < off) {
            s0[threadIdx.x] += s0[threadIdx.x + off];
            s1[threadIdx.x] += s1[threadIdx.x + off];
        }
        __syncthreads();
    }
    if (threadIdx.x == 0) {
        float mi = (s0[0] + s1[0]) * (0.5f / (float)BDIM);
        float add = ws[WS_ACC + 0] / ws[WS_ACC + 2];
        float drop = ws[WS_ACC + 1] / ws[WS_ACC + 3];
        float reg = add + drop;
        out[0] = mi + reg;   // aug_loss (ALPHA_AUG = RG_AUG = 1)
        out[1] = mi;
        out[2] = reg;
        out[3] = add;
        out[4] = drop;
    }
}

extern "C" void kernel_launch(void* const* d_in, const int* in_sizes, int n_in,
                              void* d_out, int out_size, void* d_ws, size_t ws_size,
                              hipStream_t stream) {
    const float* x    = (const float*)d_in[0];
    const float* prob = (const float*)d_in[1];
    const float* u    = (const float*)d_in[2];
    const float* z1   = (const float*)d_in[3];
    const float* z2   = (const float*)d_in[4];
    float* ws  = (float*)d_ws;
    float* out = (float*)d_out;
    int n = in_sizes[0];

    ADV_k_zero<<<64, 256, 0, stream>>>(ws, WS_POS);           // acc+row+col
    ADV_k_reg<<<4096, 256, 0, stream>>>(x, prob, u, ws, n);
    ADV_k_norm<<<(ROWS + 255) / 256, 256, 0, stream>>>(z1, z2, ws);
    ADV_k_gram<<<NST * NST / 8, 256, 0, stream>>>(ws);
    ADV_k_final<<<1, 256, 0, stream>>>(ws, out);
}